// ROIAlign_69329362092396
// MI455X (gfx1250) — compile-verified
//
#include <hip/hip_runtime.h>

typedef __attribute__((ext_vector_type(2))) float v2f;
typedef __attribute__((ext_vector_type(8))) float v8f;

namespace {
constexpr int kC = 64, kD = 48, kH = 96, kW = 96;
constexpr int kDHW = kD * kH * kW;
constexpr int kOD = 8, kOH = 8, kOW = 8;
constexpr float kScale = 0.25f;
}

// One wave32 per (roi, oh, ow). Lane = channel-in-group (0..15) x y-corner half.
// Per bin (od), per 16-channel group:
//   phase 1: issue all 32 gather loads + compute 32 weights into arrays
//   phase 2: 16 x V_WMMA_F32_16X16X4_F32 accumulate the 64 contributions
// Weights broadcast along M, channels along N; D row 0 = result.
__global__ __launch_bounds__(32)
void roialign3d_wmma_kernel(const float* __restrict__ feat,
                            const float* __restrict__ rois,
                            float* __restrict__ out)
{
  const int lane = threadIdx.x;
  const int half = lane >> 4;    // y-corner this half-wave owns (K rows 0,1 vs 2,3)
  const int cl   = lane & 15;    // channel within 16-wide group (N index)

  const int bid = blockIdx.x;    // (r*8 + oh)*8 + ow
  const int ow = bid & 7;
  const int oh = (bid >> 3) & 7;
  const int r  = bid >> 6;

  const float* roi = rois + r * 7;           // uniform per block -> scalar loads
  const int   n   = (int)roi[0];
  const float z1  = roi[1] * kScale, y1v = roi[2] * kScale, x1v = roi[3] * kScale;
  const float z2  = roi[4] * kScale, y2v = roi[5] * kScale, x2v = roi[6] * kScale;
  const float bin_d = fmaxf(z2 - z1,   1.0f) * (1.0f / kOD);
  const float bin_h = fmaxf(y2v - y1v, 1.0f) * (1.0f / kOH);
  const float bin_w = fmaxf(x2v - x1v, 1.0f) * (1.0f / kOW);

  // y / x sample prep (2 samples each, fixed for this (oh, ow))
  int   xl[2], xh[2];
  float fx[2], vy[2], vx[2];
  int   ych[2];                 // y corner for THIS half-wave, per y-sample
  float wyh[2];                 // y weight for THIS half-wave, per y-sample
#pragma unroll
  for (int i = 0; i < 2; ++i) {
    const float off = i ? 0.75f : 0.25f;

    const float cy = y1v + ((float)oh + off) * bin_h;
    vy[i] = (cy > -1.0f && cy < (float)kH) ? 1.0f : 0.0f;
    float cc = fminf(fmaxf(cy, 0.0f), (float)(kH - 1));
    float lo = floorf(cc);
    const int yli = (int)lo;
    const int yhi = (yli + 1 < kH) ? yli + 1 : kH - 1;
    const float fyi = cc - lo;
    ych[i] = half ? yhi : yli;
    wyh[i] = half ? fyi : 1.0f - fyi;

    const float cx = x1v + ((float)ow + off) * bin_w;
    vx[i] = (cx > -1.0f && cx < (float)kW) ? 1.0f : 0.0f;
    cc = fminf(fmaxf(cx, 0.0f), (float)(kW - 1));
    lo = floorf(cc);
    xl[i] = (int)lo;
    xh[i] = (xl[i] + 1 < kW) ? xl[i] + 1 : kW - 1;
    fx[i] = cc - lo;
  }

  const float* fbase = feat + (size_t)n * kC * kDHW;   // wave-uniform base

  for (int cg = 0; cg < 4; ++cg) {
    const int c     = cg * 16 + cl;
    const int coff  = c * kDHW;                        // per-lane 32-bit offset
    const int obase = (((r * kC + c) * kOH + oh) * kOW + ow) * kOD;

    for (int od = 0; od < kOD; ++od) {
      // z samples for this bin (wave-uniform, cheap)
      int   zli[2], zhi[2];
      float fzz[2], vzz[2];
#pragma unroll
      for (int iz = 0; iz < 2; ++iz) {
        const float off = iz ? 0.75f : 0.25f;
        const float cz = z1 + ((float)od + off) * bin_d;
        vzz[iz] = (cz > -1.0f && cz < (float)kD) ? 1.0f : 0.0f;
        float cc = fminf(fmaxf(cz, 0.0f), (float)(kD - 1));
        const float lo = floorf(cc);
        zli[iz] = (int)lo;
        zhi[iz] = (zli[iz] + 1 < kD) ? zli[iz] + 1 : kD - 1;
        fzz[iz] = cc - lo;
      }

      // ---- phase 1: all 32 gather loads + 32 weights into register arrays ----
      float a0[16], a1[16], b0[16], b1[16];
      int kk = 0;
#pragma unroll
      for (int iz = 0; iz < 2; ++iz) {
#pragma unroll
        for (int czk = 0; czk < 2; ++czk) {
          const float wz = czk ? fzz[iz] : 1.0f - fzz[iz];
          const int   zc = czk ? zhi[iz] : zli[iz];
#pragma unroll
          for (int iy = 0; iy < 2; ++iy) {
            const int rowb = coff + (zc * kH + ych[iy]) * kW;
            const float wzy = wz * wyh[iy] * vzz[iz] * vy[iy];
#pragma unroll
            for (int ix = 0; ix < 2; ++ix) {
              const float w = wzy * vx[ix];
              a0[kk] = w * (1.0f - fx[ix]);
              a1[kk] = w * fx[ix];
              b0[kk] = fbase[rowb + xl[ix]];
              b1[kk] = fbase[rowb + xh[ix]];
              ++kk;
            }
          }
        }
      }

      // ---- phase 2: 16 accumulating WMMAs (fp32, K=4) ----
      v8f acc = {};
#pragma unroll
      for (int k = 0; k < 16; ++k) {
        v2f A, B;
        A[0] = a0[k];  A[1] = a1[k];
        B[0] = b0[k];  B[1] = b1[k];
        acc = __builtin_amdgcn_wmma_f32_16x16x4_f32(
            false, A, false, B, (short)0, acc, false, false);
      }

      // D row 0: lanes 0..15, VGPR0 hold out[c] for this bin (x 1/8 mean)
      if (lane < 16) {
        out[obase + od] = acc[0] * 0.125f;   // layout [r][c][oh][ow][od]
      }
    }
  }
}

extern "C" void kernel_launch(void* const* d_in, const int* in_sizes, int n_in,
                              void* d_out, int out_size, void* d_ws, size_t ws_size,
                              hipStream_t stream) {
  (void)in_sizes; (void)n_in; (void)d_ws; (void)ws_size; (void)out_size;
  const float* feat = (const float*)d_in[0];   // (2, 64, 48, 96, 96) fp32
  const float* rois = (const float*)d_in[1];   // (256, 7) fp32
  float* out = (float*)d_out;                  // (256, 64, 8, 8, 8) fp32

  const int nblocks = 256 * kOH * kOW;         // one wave32 per (roi, oh, ow)
  roialign3d_wmma_kernel<<<nblocks, 32, 0, stream>>>(feat, rois, out);
}